// ContinuousAttention_87007447483016
// MI455X (gfx1250) — compile-verified
//
#include <hip/hip_runtime.h>
#include <math.h>

// Problem constants (from the reference)
#define L_LEN   2999
#define D_DIM   512
#define NB      16
#define B_BATCH 32

typedef __attribute__((ext_vector_type(2))) float v2f;
typedef __attribute__((ext_vector_type(8))) float v8f;

typedef __attribute__((ext_vector_type(4))) unsigned int u32x4;
typedef __attribute__((ext_vector_type(4))) int          i32x4;
typedef __attribute__((ext_vector_type(8))) int          i32x8;

#if defined(__has_builtin)
#  if __has_builtin(__builtin_amdgcn_tensor_load_to_lds)
#    define HAVE_TDM 1
#  endif
#endif
#ifndef HAVE_TDM
#  define HAVE_TDM 0
#endif

#define TILE_ROWS 128
#define NCHUNK ((L_LEN + TILE_ROWS - 1) / TILE_ROWS)   // 24

__device__ __forceinline__ float wave_reduce_sum(float v) {
#pragma unroll
    for (int off = 16; off > 0; off >>= 1)
        v += __shfl_xor(v, off, 32);
    return v;
}

#if HAVE_TDM
// ---------------------------------------------------------------------------
// Issue one TDM 2D tile load: rows x 16 elements of f32, row stride `stride`
// (data_size units), into LDS byte offset lds_off. tensor_dim1 = rows_rem so
// the hardware zero-fills rows past the end of the tensor (free tail handling).
// Descriptor layout per CDNA5 ISA.
// ---------------------------------------------------------------------------
__device__ __forceinline__ void tdm_load_tile(unsigned lds_off,
                                              unsigned long long gaddr,
                                              int rows_rem,
                                              unsigned stride_elems) {
    u32x4 g0;
    g0[0] = 1u;                                                 // count=1 (valid, user mode)
    g0[1] = lds_off;                                            // lds_addr (bytes)
    g0[2] = (unsigned)(gaddr & 0xFFFFFFFFull);                  // global_addr[31:0]
    g0[3] = (unsigned)((gaddr >> 32) & 0x1FFFFFFull)            // global_addr[56:32]
            | 0x80000000u;                                      // type = 2 ("image")

    unsigned rem = (unsigned)rows_rem;
    i32x8 g1;
    g1[0] = (int)0x00020000u;                                   // data_size = 2 (4 bytes)
    g1[1] = (int)(16u << 16);                                   // tensor_dim0 = 16 (lo16)
    g1[2] = (int)((rem & 0xFFFFu) << 16);                       // dim0 hi=0 | tensor_dim1 lo16
    g1[3] = (int)((16u << 16) | ((rem >> 16) & 0xFFFFu));       // tensor_dim1 hi | tile_dim0=16
    g1[4] = (int)(TILE_ROWS);                                   // tile_dim1=128 | tile_dim2=0
    g1[5] = (int)stride_elems;                                  // tensor_dim0_stride lo32
    g1[6] = 0;                                                  // stride hi | dim1_stride lo
    g1[7] = 0;                                                  // dim1_stride hi

    i32x4 z4 = {0, 0, 0, 0};
#if __clang_major__ >= 23
    i32x8 z8 = {0, 0, 0, 0, 0, 0, 0, 0};
    __builtin_amdgcn_tensor_load_to_lds(g0, g1, z4, z4, z8, 0);
#else
    __builtin_amdgcn_tensor_load_to_lds(g0, g1, z4, z4, 0);
#endif
}
#endif // HAVE_TDM

// ---------------------------------------------------------------------------
// K0: t[b,d] = sum_e W_enc[d,e] * q[b,e]
// ---------------------------------------------------------------------------
__global__ void enc_q_kernel(const float* __restrict__ q,
                             const float* __restrict__ W,
                             float* __restrict__ t) {
    int b    = blockIdx.x;
    int wv   = threadIdx.x >> 5;
    int lane = threadIdx.x & 31;

    float qv[16];
#pragma unroll
    for (int j = 0; j < 16; ++j)
        qv[j] = q[(size_t)b * D_DIM + lane + 32 * j];

    for (int d = wv; d < D_DIM; d += 8) {
        const float* wrow = W + (size_t)d * D_DIM;
        float p = 0.f;
#pragma unroll
        for (int j = 0; j < 16; ++j)
            p = fmaf(wrow[lane + 32 * j], qv[j], p);
        p = wave_reduce_sum(p);
        if (lane == 0) t[(size_t)b * D_DIM + d] = p;
    }
}

// ---------------------------------------------------------------------------
// K1: scores[b,l] = mask ? dot(keys[b,l,:], t[b,:]) / sqrt(D) : -inf
// Streams keys (196 MB) once, coalesced; prefetch runs one row-group ahead.
// ---------------------------------------------------------------------------
#define SCORE_CHUNK 256
__global__ void scores_kernel(const float* __restrict__ keys,
                              const float* __restrict__ t,
                              const unsigned char* __restrict__ mask,
                              float* __restrict__ scores) {
    int b    = blockIdx.y;
    int l0   = blockIdx.x * SCORE_CHUNK;
    int wv   = threadIdx.x >> 5;
    int lane = threadIdx.x & 31;

    float tv[16];
#pragma unroll
    for (int j = 0; j < 16; ++j)
        tv[j] = t[(size_t)b * D_DIM + lane + 32 * j];

    int lend = l0 + SCORE_CHUNK;
    if (lend > L_LEN) lend = L_LEN;
    const float scale = 0.044194173824159216f; // 1/sqrt(512)

    for (int l = l0 + wv; l < lend; l += 8) {
        const float* krow = keys + ((size_t)b * L_LEN + l) * D_DIM;
        __builtin_prefetch(krow + (size_t)8 * D_DIM + lane, 0, 1);
        float p = 0.f;
#pragma unroll
        for (int j = 0; j < 16; ++j)
            p = fmaf(krow[lane + 32 * j], tv[j], p);
        p = wave_reduce_sum(p);
        if (lane == 0) {
            float s = mask[(size_t)b * L_LEN + l] ? p * scale : -INFINITY;
            scores[(size_t)b * L_LEN + l] = s;
        }
    }
}

// ---------------------------------------------------------------------------
// K2: Bmat[b,d,n] = sum_l values[b,l,d] * G[l,n]  via V_WMMA_F32_16X16X4_F32.
// TDM double-buffers 128x16 tiles of `values` and `G` into LDS; the compute
// loop is pure ds_load + wmma (no guards; TDM zero-fills the L tail).
// 8 waves each own 16 rows of the tile; LDS tree-reduce -> deterministic.
// ---------------------------------------------------------------------------
__global__ void bmat_wmma_kernel(const float* __restrict__ values,
                                 const float* __restrict__ G,
                                 float* __restrict__ Bmat) {
    int dtile = blockIdx.x;           // 0..31 (D/16)
    int b     = blockIdx.y;           // 0..31
    int wv    = threadIdx.x >> 5;     // 0..7
    int lane  = threadIdx.x & 31;
    int half  = lane >> 4;
    int n     = lane & 15;

    v8f acc = {};

#if HAVE_TDM
    // [0,16KB): values tiles (2 bufs) | [16KB,32KB): G tiles (2 bufs) | [32KB,40KB): reduce
    __shared__ float smem[4 * TILE_ROWS * 16 + 8 * 8 * 32];
    const unsigned VOFF = 0u;
    const unsigned GOFF = (unsigned)(2 * TILE_ROWS * 16 * 4);
    const unsigned TBYTES = (unsigned)(TILE_ROWS * 16 * 4);

    unsigned long long vga = (unsigned long long)values
        + ((size_t)b * L_LEN * D_DIM + (size_t)dtile * 16) * 4ull;
    unsigned long long gga = (unsigned long long)G;

    if (wv == 0) {
        tdm_load_tile(VOFF, vga, L_LEN, D_DIM);
        tdm_load_tile(GOFF, gga, L_LEN, NB);
        __builtin_amdgcn_s_wait_tensorcnt(0);
    }
    __syncthreads();

    for (int c = 0; c < NCHUNK; ++c) {
        int cur = c & 1;
        if (wv == 0 && (c + 1) < NCHUNK) {
            unsigned nxt = (unsigned)((c + 1) & 1);
            int rem = L_LEN - (c + 1) * TILE_ROWS;
            tdm_load_tile(VOFF + nxt * TBYTES,
                          vga + (unsigned long long)(c + 1) * TILE_ROWS * D_DIM * 4ull,
                          rem, D_DIM);
            tdm_load_tile(GOFF + nxt * TBYTES,
                          gga + (unsigned long long)(c + 1) * TILE_ROWS * NB * 4ull,
                          rem, NB);
        }

        const float* vt = smem + cur * (TILE_ROWS * 16);
        const float* gt = smem + 2 * TILE_ROWS * 16 + cur * (TILE_ROWS * 16);
        int rb = wv * 16;                       // this wave's 16 rows of the tile
#pragma unroll
        for (int u = 0; u < 4; ++u) {
            int k0 = rb + u * 4 + 2 * half;
            v2f a, bb;
            a.x  = vt[k0 * 16 + n];
            a.y  = vt[(k0 + 1) * 16 + n];
            bb.x = gt[k0 * 16 + n];
            bb.y = gt[(k0 + 1) * 16 + n];
            acc = __builtin_amdgcn_wmma_f32_16x16x4_f32(
                false, a, false, bb, (short)0, acc, false, false);
        }

        if (wv == 0 && (c + 1) < NCHUNK)
            __builtin_amdgcn_s_wait_tensorcnt(0);   // next buffers landed
        __syncthreads();                            // release buffers / publish
    }
    float* red = smem + 4 * TILE_ROWS * 16;
#else
    // Fallback: direct global loads, guard-free main body + one guarded tail.
    __shared__ float redbuf[8 * 8 * 32];
    float* red = redbuf;
    const float* vbase = values + (size_t)b * L_LEN * D_DIM;
    int d = dtile * 16 + n;
    int l;
    for (l = wv * 8; l + 8 <= L_LEN; l += 64) {
        __builtin_prefetch(vbase + (size_t)(l + 64) * D_DIM + d, 0, 1);
#pragma unroll
        for (int u = 0; u < 2; ++u) {
            int k0 = l + u * 4 + 2 * half;
            v2f a, bb;
            a.x  = vbase[(size_t)k0 * D_DIM + d];
            a.y  = vbase[(size_t)(k0 + 1) * D_DIM + d];
            bb.x = G[(size_t)k0 * NB + n];
            bb.y = G[(size_t)(k0 + 1) * NB + n];
            acc = __builtin_amdgcn_wmma_f32_16x16x4_f32(
                false, a, false, bb, (short)0, acc, false, false);
        }
    }
    if (l < L_LEN) {
#pragma unroll
        for (int u = 0; u < 2; ++u) {
            int k0 = l + u * 4 + 2 * half;
            int k1 = k0 + 1;
            v2f a, bb;
            a.x  = (k0 < L_LEN) ? vbase[(size_t)k0 * D_DIM + d] : 0.f;
            a.y  = (k1 < L_LEN) ? vbase[(size_t)k1 * D_DIM + d] : 0.f;
            bb.x = (k0 < L_LEN) ? G[(size_t)k0 * NB + n] : 0.f;
            bb.y = (k1 < L_LEN) ? G[(size_t)k1 * NB + n] : 0.f;
            acc = __builtin_amdgcn_wmma_f32_16x16x4_f32(
                false, a, false, bb, (short)0, acc, false, false);
        }
    }
#endif

    // cross-wave reduction of the 8 partial 16x16 accumulators
#pragma unroll
    for (int i = 0; i < 8; ++i)
        red[(wv * 8 + i) * 32 + lane] = acc[i];
    __syncthreads();

    int i = threadIdx.x >> 5;
    float s = 0.f;
#pragma unroll
    for (int w2 = 0; w2 < 8; ++w2)
        s += red[(w2 * 8 + i) * 32 + lane];
    int M = (lane < 16) ? i : (i + 8);      // C/D layout: vgpr i, lane half
    Bmat[((size_t)b * D_DIM + dtile * 16 + M) * NB + (lane & 15)] = s;
}

// ---------------------------------------------------------------------------
// K3: per batch: softmax stats over scores -> mu, var -> r[16]; c = Bmat @ r
// ---------------------------------------------------------------------------
__global__ void finalize_kernel(const float* __restrict__ scores,
                                const float* __restrict__ Bmat,
                                const float* __restrict__ bmu,
                                const float* __restrict__ bsig,
                                float* __restrict__ out) {
    int b   = blockIdx.x;
    int tid = threadIdx.x;
    const float* sc = scores + (size_t)b * L_LEN;

    __shared__ float red0[256];
    __shared__ float red1[256];
    __shared__ float red2[256];

    float m = -INFINITY;
    for (int l = tid; l < L_LEN; l += 256) m = fmaxf(m, sc[l]);
    red0[tid] = m;
    __syncthreads();
    for (int s = 128; s > 0; s >>= 1) {
        if (tid < s) red0[tid] = fmaxf(red0[tid], red0[tid + s]);
        __syncthreads();
    }
    float mx = red0[0];
    __syncthreads();

    const float invL = 1.0f / (float)L_LEN;
    float s0 = 0.f, s1 = 0.f, s2 = 0.f;
    for (int l = tid; l < L_LEN; l += 256) {
        float e = __expf(sc[l] - mx);
        float p = ((float)l + 0.5f) * invL;   // pos[l] = (l+0.5)/L
        s0 += e;
        s1 += e * p;
        s2 += e * p * p;
    }
    red0[tid] = s0; red1[tid] = s1; red2[tid] = s2;
    __syncthreads();
    for (int s = 128; s > 0; s >>= 1) {
        if (tid < s) {
            red0[tid] += red0[tid + s];
            red1[tid] += red1[tid + s];
            red2[tid] += red2[tid + s];
        }
        __syncthreads();
    }

    float inv0 = 1.0f / red0[0];
    float mu   = red1[0] * inv0;
    float e2   = red2[0] * inv0;
    float var  = fmaxf(e2 - mu * mu, 1e-7f);

    __shared__ float rsh[NB];
    if (tid < NB) {
        float ts = var + bsig[tid] * bsig[tid];
        float dm = mu - bmu[tid];
        rsh[tid] = 0.3989422804014327f * rsqrtf(ts) * __expf(-0.5f * dm * dm / ts);
    }
    __syncthreads();

    const float* Bb = Bmat + (size_t)b * D_DIM * NB;
    for (int d = tid; d < D_DIM; d += 256) {
        float c = 0.f;
#pragma unroll
        for (int nn = 0; nn < NB; ++nn)
            c = fmaf(Bb[d * NB + nn], rsh[nn], c);
        out[(size_t)b * D_DIM + d] = c;
    }
}

// ---------------------------------------------------------------------------
extern "C" void kernel_launch(void* const* d_in, const int* in_sizes, int n_in,
                              void* d_out, int out_size, void* d_ws, size_t ws_size,
                              hipStream_t stream) {
    const float* query        = (const float*)d_in[0];         // (B,1,D)
    const float* keys         = (const float*)d_in[1];         // (B,L,D)
    const float* values       = (const float*)d_in[2];         // (B,L,D)
    const unsigned char* mask = (const unsigned char*)d_in[3]; // (B,L) bool
    const float* W_enc        = (const float*)d_in[4];         // (D,D)
    const float* G            = (const float*)d_in[5];         // (L,NB)
    const float* bmu          = (const float*)d_in[6];         // (NB)
    const float* bsig         = (const float*)d_in[7];         // (NB)
    float* out = (float*)d_out;                                // (B,1,D)

    float* ws     = (float*)d_ws;
    float* t      = ws;                            // B*D
    float* scores = t + B_BATCH * D_DIM;           // B*L
    float* Bmat   = scores + B_BATCH * L_LEN;      // B*D*NB

    enc_q_kernel<<<B_BATCH, 256, 0, stream>>>(query, W_enc, t);

    dim3 g1((L_LEN + SCORE_CHUNK - 1) / SCORE_CHUNK, B_BATCH);
    scores_kernel<<<g1, 256, 0, stream>>>(keys, t, mask, scores);

    dim3 g2(D_DIM / 16, B_BATCH);
    bmat_wmma_kernel<<<g2, 256, 0, stream>>>(values, G, Bmat);

    finalize_kernel<<<B_BATCH, 256, 0, stream>>>(scores, Bmat, bmu, bsig, out);
}